// GCN_55284819034515
// MI455X (gfx1250) — compile-verified
//
#include <hip/hip_runtime.h>

typedef float v2f __attribute__((ext_vector_type(2)));
typedef float v8f __attribute__((ext_vector_type(8)));

#define FEAT 128
#define CLS  40

__device__ __forceinline__ unsigned mix32(unsigned x) {
  x ^= x >> 16; x *= 0x7feb352dU;
  x ^= x >> 15; x *= 0x846ca68bU;
  x ^= x >> 16;
  return x;
}

// ---------------- degree / normalization ----------------
__global__ void k_init_deg(float* __restrict__ deg, int n) {
  int i = blockIdx.x * blockDim.x + threadIdx.x;
  if (i < n) deg[i] = 1.0f;  // self-loop contributes 1 to every node's degree
}

__global__ void k_accum_deg(const int* __restrict__ dst, float* __restrict__ deg, int e) {
  int i = blockIdx.x * blockDim.x + threadIdx.x;
  if (i < e) unsafeAtomicAdd(&deg[dst[i]], 1.0f);
}

__global__ void k_rsqrt_inplace(float* __restrict__ deg, int n) {
  int i = blockIdx.x * blockDim.x + threadIdx.x;
  if (i < n) deg[i] = rsqrtf(deg[i]);
}

__global__ void k_edge_norm(const int* __restrict__ src, const int* __restrict__ dst,
                            const float* __restrict__ dinv, float* __restrict__ nrm, int e) {
  int i = blockIdx.x * blockDim.x + threadIdx.x;
  if (i < e) nrm[i] = dinv[src[i]] * dinv[dst[i]];
}

// ---------------- GEMM1: H = X (n x 128) @ W1 (128 x 128), fp32 WMMA ----------------
// One block = 256 threads = 8 waves; each wave owns one 16x16 tile of the
// 16 x 128 output row-slab. K marched 4 at a time with V_WMMA_F32_16X16X4_F32.
__global__ __launch_bounds__(256) void k_gemm1(const float* __restrict__ X,
                                               const float* __restrict__ W,
                                               float* __restrict__ H, int n) {
  __shared__ float As[16 * 132];  // stride 132 to avoid LDS bank conflicts
  const int m0 = blockIdx.x << 4;
  for (int i = threadIdx.x; i < 16 * FEAT; i += 256) {
    int r = i >> 7, c = i & 127;
    int gr = m0 + r;
    As[r * 132 + c] = (gr < n) ? X[(long long)gr * FEAT + c] : 0.0f;
  }
  __syncthreads();

  const int lane = threadIdx.x & 31;
  const int n0   = (threadIdx.x >> 5) << 4;  // wave -> column tile (0..112)
  const int m    = lane & 15;
  const int half = lane >> 4;                // 0: K+0/K+1, 1: K+2/K+3 (ISA A/B layout)
  const int koff = half << 1;
  const int ncol = n0 + m;

  v8f acc = {};
#pragma unroll
  for (int k0 = 0; k0 < FEAT; k0 += 4) {
    v2f a, b;
    a.x = As[m * 132 + k0 + koff];
    a.y = As[m * 132 + k0 + koff + 1];
    b.x = W[(k0 + koff) * FEAT + ncol];
    b.y = W[(k0 + koff + 1) * FEAT + ncol];
    acc = __builtin_amdgcn_wmma_f32_16x16x4_f32(false, a, false, b, (short)0, acc,
                                                false, false);
  }
  // C layout: VGPR v -> row m0+v (lanes 0-15) / m0+v+8 (lanes 16-31), col n0+(lane&15)
#pragma unroll
  for (int v = 0; v < 8; ++v) {
    int row = m0 + v + (half << 3);
    if (row < n) H[(long long)row * FEAT + ncol] = acc[v];
  }
}

// ---------------- GEMM2: H2 = Hd (n x 128) @ W2 (128 x 40), fp32 WMMA ----------------
// 192 threads = 6 waves: 2 row-tiles x 3 col-tiles (48 cols covers 40 classes).
__global__ __launch_bounds__(192) void k_gemm2(const float* __restrict__ Hin,
                                               const float* __restrict__ W2,
                                               float* __restrict__ H2, int n) {
  __shared__ float As[32 * 132];
  const int m0 = blockIdx.x << 5;
  for (int i = threadIdx.x; i < 32 * FEAT; i += 192) {
    int r = i >> 7, c = i & 127;
    int gr = m0 + r;
    As[r * 132 + c] = (gr < n) ? Hin[(long long)gr * FEAT + c] : 0.0f;
  }
  __syncthreads();

  const int wave = threadIdx.x >> 5;   // 0..5
  const int lane = threadIdx.x & 31;
  const int rt   = wave / 3;           // row tile 0..1
  const int n0   = (wave % 3) << 4;    // col tile 0,16,32
  const int m    = (rt << 4) + (lane & 15);
  const int half = lane >> 4;
  const int koff = half << 1;
  const int ncol = n0 + (lane & 15);
  const bool colok = (ncol < CLS);

  v8f acc = {};
#pragma unroll
  for (int k0 = 0; k0 < FEAT; k0 += 4) {
    v2f a, b;
    a.x = As[m * 132 + k0 + koff];
    a.y = As[m * 132 + k0 + koff + 1];
    b.x = colok ? W2[(k0 + koff) * CLS + ncol] : 0.0f;
    b.y = colok ? W2[(k0 + koff + 1) * CLS + ncol] : 0.0f;
    acc = __builtin_amdgcn_wmma_f32_16x16x4_f32(false, a, false, b, (short)0, acc,
                                                false, false);
  }
#pragma unroll
  for (int v = 0; v < 8; ++v) {
    int row = m0 + (rt << 4) + v + (half << 3);
    if (row < n && colok) H2[(long long)row * CLS + ncol] = acc[v];
  }
}

// ---------------- layer-1 aggregation ----------------
// init accumulator with the self-loop term: out[n] = H[n] * dinv[n]^2
__global__ void k_self1(const float* __restrict__ H, const float* __restrict__ dinv,
                        float* __restrict__ out, long long nwork) {
  long long idx = (long long)blockIdx.x * blockDim.x + threadIdx.x;
  if (idx >= nwork) return;
  int node = (int)(idx >> 5);
  int f4 = (int)(idx & 31) << 2;
  float s = dinv[node];
  s *= s;
  float4 h = *(const float4*)(H + (long long)node * FEAT + f4);
  float4 o = make_float4(h.x * s, h.y * s, h.z * s, h.w * s);
  *(float4*)(out + (long long)node * FEAT + f4) = o;
}

// edge scatter: out[dst] += H[src] * norm[e]   (hardware fp32 atomics, L2-resident)
__global__ void k_edge1(const float* __restrict__ H, const int* __restrict__ src,
                        const int* __restrict__ dst, const float* __restrict__ nrm,
                        float* __restrict__ out, long long nwork) {
  long long idx = (long long)blockIdx.x * blockDim.x + threadIdx.x;
  if (idx >= nwork) return;
  int e  = (int)(idx >> 5);
  int f4 = (int)(idx & 31) << 2;
  float w = nrm[e];
  const float4 h = *(const float4*)(H + (long long)src[e] * FEAT + f4);
  float* o = out + (long long)dst[e] * FEAT + f4;
  unsafeAtomicAdd(o + 0, h.x * w);
  unsafeAtomicAdd(o + 1, h.y * w);
  unsafeAtomicAdd(o + 2, h.z * w);
  unsafeAtomicAdd(o + 3, h.w * w);
}

// bias + ReLU + dropout(p=0.5) in place (stateless hash mask, scale 2x on keep)
__global__ void k_postact(float* __restrict__ buf, const float* __restrict__ b1,
                          long long nwork) {
  long long idx = (long long)blockIdx.x * blockDim.x + threadIdx.x;
  if (idx >= nwork) return;
  float v = buf[idx] + b1[(int)(idx & 127)];
  v = fmaxf(v, 0.0f);
  unsigned hsh = mix32((unsigned)idx ^ 0x2545F491u);
  buf[idx] = (hsh >> 31) ? v * 2.0f : 0.0f;
}

// ---------------- layer-2 aggregation ----------------
// d_out init: self-loop term + bias (fully rewrites d_out -> replay-safe)
__global__ void k_self2(const float* __restrict__ H2, const float* __restrict__ dinv,
                        const float* __restrict__ b2, float* __restrict__ out,
                        long long nwork) {
  long long idx = (long long)blockIdx.x * blockDim.x + threadIdx.x;
  if (idx >= nwork) return;
  int node = (int)(idx / CLS);
  int c = (int)(idx - (long long)node * CLS);
  float s = dinv[node];
  s *= s;
  out[idx] = H2[idx] * s + b2[c];
}

__global__ void k_edge2(const float* __restrict__ H2, const int* __restrict__ src,
                        const int* __restrict__ dst, const float* __restrict__ nrm,
                        float* __restrict__ out, long long nwork) {
  long long idx = (long long)blockIdx.x * blockDim.x + threadIdx.x;
  if (idx >= nwork) return;
  int e  = (int)(idx / 10);
  int c4 = (int)(idx - (long long)e * 10) << 2;  // 10 float4 chunks cover 40 classes
  float w = nrm[e];
  const float4 h = *(const float4*)(H2 + (long long)src[e] * CLS + c4);
  float* o = out + (long long)dst[e] * CLS + c4;
  unsafeAtomicAdd(o + 0, h.x * w);
  unsafeAtomicAdd(o + 1, h.y * w);
  unsafeAtomicAdd(o + 2, h.z * w);
  unsafeAtomicAdd(o + 3, h.w * w);
}

// ---------------- host launch ----------------
extern "C" void kernel_launch(void* const* d_in, const int* in_sizes, int n_in,
                              void* d_out, int out_size, void* d_ws, size_t ws_size,
                              hipStream_t stream) {
  (void)n_in; (void)out_size; (void)ws_size;
  const float* x  = (const float*)d_in[0];
  const int*   ei = (const int*)d_in[1];
  const float* W1 = (const float*)d_in[2];
  const float* b1 = (const float*)d_in[3];
  const float* W2 = (const float*)d_in[4];
  const float* b2 = (const float*)d_in[5];
  float* out = (float*)d_out;

  const int n = in_sizes[0] / FEAT;   // 100000
  const int e = in_sizes[1] / 2;      // 1600000
  const int* srcv = ei;
  const int* dstv = ei + e;

  // workspace carve-up (512B aligned): dinv | edge-norm | bufA (n*128) | bufB (n*128)
  char* w = (char*)d_ws;
  size_t off = 0;
  float* dinv = (float*)(w + off); off += (((size_t)n * 4) + 511) & ~(size_t)511;
  float* nrm  = (float*)(w + off); off += (((size_t)e * 4) + 511) & ~(size_t)511;
  float* bufA = (float*)(w + off); off += (((size_t)n * FEAT * 4) + 511) & ~(size_t)511;
  float* bufB = (float*)(w + off);

  const int T = 256;
  auto nb = [](long long work, int t) { return (unsigned)((work + t - 1) / t); };

  k_init_deg     <<<nb(n, T), T, 0, stream>>>(dinv, n);
  k_accum_deg    <<<nb(e, T), T, 0, stream>>>(dstv, dinv, e);
  k_rsqrt_inplace<<<nb(n, T), T, 0, stream>>>(dinv, n);
  k_edge_norm    <<<nb(e, T), T, 0, stream>>>(srcv, dstv, dinv, nrm, e);

  k_gemm1<<<(unsigned)((n + 15) / 16), 256, 0, stream>>>(x, W1, bufA, n);

  long long w_self1 = (long long)n * 32;   // n * (128/4)
  k_self1<<<nb(w_self1, T), T, 0, stream>>>(bufA, dinv, bufB, w_self1);
  long long w_edge1 = (long long)e * 32;   // e * (128/4)
  k_edge1<<<nb(w_edge1, T), T, 0, stream>>>(bufA, srcv, dstv, nrm, bufB, w_edge1);

  long long w_post = (long long)n * FEAT;
  k_postact<<<nb(w_post, T), T, 0, stream>>>(bufB, b1, w_post);

  k_gemm2<<<(unsigned)((n + 31) / 32), 192, 0, stream>>>(bufB, W2, bufA, n);

  long long w_self2 = (long long)n * CLS;
  k_self2<<<nb(w_self2, T), T, 0, stream>>>(bufA, dinv, b2, out, w_self2);
  long long w_edge2 = (long long)e * 10;   // e * (40/4)
  k_edge2<<<nb(w_edge2, T), T, 0, stream>>>(bufA, srcv, dstv, nrm, out, w_edge2);
}